// ReservoirLayer_29875792511773
// MI455X (gfx1250) — compile-verified
//
#include <hip/hip_runtime.h>
#include <cstddef>
#include <cstdint>

typedef __attribute__((ext_vector_type(2))) float v2f;
typedef __attribute__((ext_vector_type(8))) float v8f;

#define SEQ    2048
#define NB     16      // batch
#define DMODEL 1024
#define NRES   1024
#define LEAK   0.3f
#define KEEP   0.7f
#define NWG    32              // persistent scan workgroups
#define RW     (NRES / NWG)    // 32 reservoir columns per scan WG
#define WSTRIDE 1028           // padded LDS row stride (floats): 16B-aligned rows,
                               // stride%64==4 -> conflict-free WMMA reads
#define GSTRIDE 68             // GEMM LDS panel stride (floats): same properties

// ---------------------------------------------------------------------------
// CDNA5 async global->LDS copy (ASYNCcnt path), via inline asm so it works on
// both toolchains regardless of builtin arity. One call moves 16B per lane.
// ---------------------------------------------------------------------------
__device__ __forceinline__ uint32_t lds_off(const void* p) {
  // flat LDS aperture keeps the wave-relative LDS byte offset in the low 32 bits
  return (uint32_t)(uintptr_t)p;
}
__device__ __forceinline__ void async_ld_b128(uint32_t lds_addr, const void* g) {
  asm volatile("global_load_async_to_lds_b128 %0, %1, off"
               :: "v"(lds_addr), "v"((uint64_t)(uintptr_t)g) : "memory");
}
__device__ __forceinline__ void async_ld_b128_dev(uint32_t lds_addr, const void* g) {
  // device-scope: state slices were produced by other WGPs last step
  asm volatile("global_load_async_to_lds_b128 %0, %1, off scope:SCOPE_DEV"
               :: "v"(lds_addr), "v"((uint64_t)(uintptr_t)g) : "memory");
}
__device__ __forceinline__ void wait_async0() {
  asm volatile("s_wait_asynccnt 0x0" ::: "memory");
}

// ---------------------------------------------------------------------------
// One 16x16 fp32 WMMA tile, K-chunk of 64; A/B both read as row[am]*stride+k+ak
// (B staged n-major so its lane addressing matches A's).
// ---------------------------------------------------------------------------
__device__ __forceinline__ v8f wmma_k64(const float* arow, const float* brow,
                                        int ak, v8f acc) {
#pragma unroll
  for (int k = 0; k < 64; k += 4) {
    v2f a, b;
    a.x = arow[k + ak]; a.y = arow[k + ak + 1];
    b.x = brow[k + ak]; b.y = brow[k + ak + 1];
    acc = __builtin_amdgcn_wmma_f32_16x16x4_f32(false, a, false, b,
                                                (short)0, acc, false, false);
  }
  return acc;
}

// ---------------------------------------------------------------------------
// Kernel A: per-row LayerNorm statistics. One wave32 per (b,s) row, m = s*16+b.
// ---------------------------------------------------------------------------
__global__ __launch_bounds__(256) void ln_stats_kernel(const float* __restrict__ x,
                                                       float* __restrict__ mu,
                                                       float* __restrict__ rstd) {
  const int wid  = (blockIdx.x * 256 + threadIdx.x) >> 5;  // row m = s*16+b
  const int lane = threadIdx.x & 31;
  const int s = wid >> 4, b = wid & 15;
  const float* row = x + ((size_t)b * SEQ + s) * DMODEL;
  float sum = 0.f, sq = 0.f;
#pragma unroll
  for (int i = 0; i < 8; ++i) {
    float4 v = *(const float4*)(row + i * 128 + lane * 4);
    sum += v.x + v.y + v.z + v.w;
    sq  += v.x * v.x + v.y * v.y + v.z * v.z + v.w * v.w;
  }
#pragma unroll
  for (int off = 16; off; off >>= 1) {
    sum += __shfl_xor(sum, off, 32);
    sq  += __shfl_xor(sq,  off, 32);
  }
  if (lane == 0) {
    float m   = sum * (1.0f / DMODEL);
    float var = sq  * (1.0f / DMODEL) - m * m;
    mu[wid]   = m;
    rstd[wid] = rsqrtf(var + 1e-5f);
  }
}

// ---------------------------------------------------------------------------
// Kernel B: inp[s][b][r] = LN(x)[b][s][:] . W_in[r][:]   (M=32768,N=1024,K=1024)
// Block = 16(M) x 64(N) tile, 4 waves (one 16x16 WMMA tile each), K-chunk 64.
// LayerNorm fused into the A staging; B panel staged via async global->LDS.
// ---------------------------------------------------------------------------
__global__ __launch_bounds__(128) void inp_gemm_kernel(
    const float* __restrict__ x, const float* __restrict__ W_in,
    const float* __restrict__ gamma, const float* __restrict__ beta,
    const float* __restrict__ mu, const float* __restrict__ rstd,
    float* __restrict__ inp) {
  __shared__ float As[16 * GSTRIDE];
  __shared__ float Bs[64 * GSTRIDE];
  const int s = blockIdx.x;        // M tile == one sequence index (rows are b)
  const int nblk = blockIdx.y;
  const int tid = threadIdx.x, lane = tid & 31, w = tid >> 5;
  const int am = lane & 15, ak = (lane >> 4) << 1;
  v8f acc = {};
  for (int kc = 0; kc < DMODEL; kc += 64) {
    __syncthreads();
    // stage B async, n-major: Bs[n][k] = W_in[nblk*64+n][kc+k]
    for (int q = tid; q < 1024; q += 128) {
      int n = q >> 4, kv = (q & 15) << 2;
      async_ld_b128(lds_off(Bs + n * GSTRIDE + kv),
                    W_in + (size_t)(nblk * 64 + n) * DMODEL + kc + kv);
    }
    // stage A manually (LayerNorm fused), rows b=0..15
    for (int q = tid; q < 256; q += 128) {
      int r = q >> 4, kv = (q & 15) << 2, k = kc + kv;
      float4 v = *(const float4*)(x + ((size_t)r * SEQ + s) * DMODEL + k);
      float mm = mu[s * 16 + r], rs = rstd[s * 16 + r];
      float* d = As + r * GSTRIDE + kv;
      d[0] = (v.x - mm) * rs * gamma[k]     + beta[k];
      d[1] = (v.y - mm) * rs * gamma[k + 1] + beta[k + 1];
      d[2] = (v.z - mm) * rs * gamma[k + 2] + beta[k + 2];
      d[3] = (v.w - mm) * rs * gamma[k + 3] + beta[k + 3];
    }
    wait_async0();
    __syncthreads();
    acc = wmma_k64(As + am * GSTRIDE, Bs + (w * 16 + am) * GSTRIDE, ak, acc);
  }
  const int gcol = nblk * 64 + w * 16 + am;
  const int half = (lane >> 4) << 3;
#pragma unroll
  for (int j = 0; j < 8; ++j) {
    int mrow = j + half;
    inp[((size_t)s * 16 + mrow) * NRES + gcol] = acc[j];   // (S,B,R) layout
  }
}

// ---------------------------------------------------------------------------
// Kernel C: serial reservoir scan. 32 persistent WGs x 64 threads (2 waves).
// Each WG pins its 32-column W_res^T panel in LDS (~132KB) for all 2048 steps,
// refreshes the 16x1024 state from a small global buffer each step via async
// global->LDS (device scope), computes its 16x32 slice with fp32 WMMA, applies
// leak/tanh, and syncs all WGs with a per-step atomic counter barrier.
// states[t] overwrites inp[t] in place.
// ---------------------------------------------------------------------------
__global__ __launch_bounds__(64) void scan_kernel(
    float* __restrict__ buf,            // in: inp (S,B,R); out: states (S,B,R)
    const float* __restrict__ W_res,
    float* __restrict__ Sg,             // 16x1024 global state (pre-zeroed)
    int* __restrict__ cnt) {            // SEQ counters (pre-zeroed)
  extern __shared__ float lds[];
  float* Ws = lds;                      // RW x WSTRIDE  (W_res panel, n-major)
  float* Ss = lds + RW * WSTRIDE;       // 16 x WSTRIDE  (full state)
  const int wg = blockIdx.x, tid = threadIdx.x, lane = tid & 31, w = tid >> 5;
  const int r0 = wg * RW;
  const int am = lane & 15, ak = (lane >> 4) << 1;

  // Pin W_res panel (async): Ws[n][k] = W_res[r0+n][k]
  for (int q = tid; q < RW * (NRES / 4); q += 64) {
    int n = q >> 8, kv = (q & 255) << 2;
    async_ld_b128(lds_off(Ws + n * WSTRIDE + kv),
                  W_res + (size_t)(r0 + n) * NRES + kv);
  }
  const float* arow = Ss + am * WSTRIDE;
  const float* brow = Ws + (w * 16 + am) * WSTRIDE;
  const int gcol = r0 + w * 16 + am;
  const int half = (lane >> 4) << 3;

  for (int t = 0; t < SEQ; ++t) {
    __syncthreads();                    // prior reads of Ss done before overwrite
    // refresh full state into LDS (async, device scope; also covers Ws at t=0)
    for (int q = tid; q < 16 * (NRES / 4); q += 64) {
      int m = q >> 8, kv = (q & 255) << 2;
      async_ld_b128_dev(lds_off(Ss + m * WSTRIDE + kv), Sg + m * NRES + kv);
    }
    wait_async0();
    __syncthreads();
    if (t + 1 < SEQ)  // prefetch next step's input drive slice
      __builtin_prefetch(buf + ((size_t)(t + 1) * NB + (tid & 15)) * NRES + r0, 0, 0);

    v8f acc = {};
#pragma unroll 16
    for (int k = 0; k < NRES; k += 4) {
      v2f a, b;
      a.x = arow[k + ak]; a.y = arow[k + ak + 1];
      b.x = brow[k + ak]; b.y = brow[k + ak + 1];
      acc = __builtin_amdgcn_wmma_f32_16x16x4_f32(false, a, false, b,
                                                  (short)0, acc, false, false);
    }
#pragma unroll
    for (int j = 0; j < 8; ++j) {
      int m = j + half;
      size_t idx = ((size_t)t * NB + m) * NRES + gcol;
      float u  = buf[idx];
      float nv = KEEP * Ss[m * WSTRIDE + gcol] + LEAK * tanhf(acc[j] + u);
      buf[idx] = nv;                 // states[t] (in place over inp[t])
      Sg[m * NRES + gcol] = nv;      // publish state slice
    }
    // device-wide step barrier
    __threadfence();
    __syncthreads();
    if (tid == 0) {
      __hip_atomic_fetch_add(&cnt[t], 1, __ATOMIC_ACQ_REL, __HIP_MEMORY_SCOPE_AGENT);
      while (__hip_atomic_load(&cnt[t], __ATOMIC_ACQUIRE, __HIP_MEMORY_SCOPE_AGENT) < NWG)
        __builtin_amdgcn_s_sleep(1);
    }
    __syncthreads();
    __threadfence();
  }
}

// ---------------------------------------------------------------------------
// Kernel D: out[b][s][d] = x[b][s][d] + states[s][b][:] . ro_w[d][:] + ro_b[d]
// Same tiling as kernel B; both panels staged async; residual+bias in epilogue.
// ---------------------------------------------------------------------------
__global__ __launch_bounds__(128) void readout_gemm_kernel(
    const float* __restrict__ states, const float* __restrict__ ro_w,
    const float* __restrict__ ro_b, const float* __restrict__ x,
    float* __restrict__ out) {
  __shared__ float As[16 * GSTRIDE];
  __shared__ float Bs[64 * GSTRIDE];
  const int s = blockIdx.x, nblk = blockIdx.y;
  const int tid = threadIdx.x, lane = tid & 31, w = tid >> 5;
  const int am = lane & 15, ak = (lane >> 4) << 1;
  v8f acc = {};
  for (int kc = 0; kc < NRES; kc += 64) {
    __syncthreads();
    for (int q = tid; q < 256; q += 128) {
      int r = q >> 4, kv = (q & 15) << 2;
      async_ld_b128(lds_off(As + r * GSTRIDE + kv),
                    states + ((size_t)s * 16 + r) * NRES + kc + kv);
    }
    for (int q = tid; q < 1024; q += 128) {
      int n = q >> 4, kv = (q & 15) << 2;
      async_ld_b128(lds_off(Bs + n * GSTRIDE + kv),
                    ro_w + (size_t)(nblk * 64 + n) * NRES + kc + kv);
    }
    wait_async0();
    __syncthreads();
    acc = wmma_k64(As + am * GSTRIDE, Bs + (w * 16 + am) * GSTRIDE, ak, acc);
  }
  const int gcol = nblk * 64 + w * 16 + am;
  const float bias = ro_b[gcol];
  const int half = (lane >> 4) << 3;
#pragma unroll
  for (int j = 0; j < 8; ++j) {
    int b = j + half;
    size_t oidx = ((size_t)b * SEQ + s) * DMODEL + gcol;
    out[oidx] = x[oidx] + acc[j] + bias;
  }
}

// ---------------------------------------------------------------------------
extern "C" void kernel_launch(void* const* d_in, const int* in_sizes, int n_in,
                              void* d_out, int out_size, void* d_ws, size_t ws_size,
                              hipStream_t stream) {
  (void)in_sizes; (void)n_in; (void)out_size; (void)ws_size;
  const float* x     = (const float*)d_in[0];
  const float* W_res = (const float*)d_in[1];
  const float* W_in  = (const float*)d_in[2];
  const float* ro_w  = (const float*)d_in[3];
  const float* ro_b  = (const float*)d_in[4];
  const float* gamma = (const float*)d_in[5];
  const float* beta  = (const float*)d_in[6];
  float* out = (float*)d_out;

  // Workspace layout
  float* mu   = (float*)d_ws;                 // 32768 floats
  float* rstd = mu + 32768;                   // 32768 floats
  float* Sg   = rstd + 32768;                 // 16384 floats (16x1024 state)
  int*   cnt  = (int*)(Sg + 16384);           // 2048 ints (per-step barrier)
  float* buf  = (float*)((char*)d_ws + 335872); // 128MB: inp -> states in place

  // zero the state + barrier counters every launch (graph-capture safe)
  hipMemsetAsync(Sg, 0, (16384 + 2048) * 4, stream);

  ln_stats_kernel<<<dim3(4096), 256, 0, stream>>>(x, mu, rstd);
  inp_gemm_kernel<<<dim3(SEQ, 16), 128, 0, stream>>>(x, W_in, gamma, beta, mu, rstd, buf);
  scan_kernel<<<dim3(NWG), 64, (RW + 16) * WSTRIDE * sizeof(float), stream>>>(buf, W_res, Sg, cnt);
  readout_gemm_kernel<<<dim3(SEQ, 16), 128, 0, stream>>>(buf, ro_w, ro_b, x, out);
}